// Model_6219112645415
// MI455X (gfx1250) — compile-verified
//
#include <hip/hip_runtime.h>
#include <stdint.h>

// ---------------- configuration ----------------
#define CHUNK 128          // edges per wave per DMA chunk
#define WAVES 8            // waves per 256-thread block (wave32)
#define BLOCK 256

// Real spherical harmonic constants (l = 0,1,2)
#define SH_C0  0.28209479177387814f
#define SH_C1  0.4886025119029199f
#define SH_C2A 1.0925484305920792f
#define SH_C2B 0.31539156525252005f
#define SH_C2C 0.5462742152960396f

typedef unsigned int v4u __attribute__((ext_vector_type(4)));
typedef int          v8i __attribute__((ext_vector_type(8)));
typedef int          v4i __attribute__((ext_vector_type(4)));

#if defined(__has_builtin)
#if __has_builtin(__builtin_amdgcn_tensor_load_to_lds) && \
    __has_builtin(__builtin_amdgcn_s_wait_tensorcnt)
#define USE_TDM 1
#endif
#endif
#ifndef USE_TDM
#define USE_TDM 0
#endif

// ---------------- TDM: 1-D DMA of `tile_elems` dwords global -> LDS ----------------
// D# per CDNA5 ISA:
//   group0: count=1, lds_addr, global_addr[56:0], type=2
//   group1: data_size=4B, tensor_dim0 = #valid elements (OOB reads return 0),
//           tensor_dim1=1, tile_dim0 = tile_elems (1-D tile: tile_dim1=0)
__device__ __forceinline__ void tdm_load_1d(uint32_t lds_byte_addr, const void* gptr,
                                            uint32_t tensor_elems, uint32_t tile_elems) {
#if USE_TDM
  uint64_t ga = (uint64_t)(uintptr_t)gptr;
  v4u g0;
  g0[0] = 1u;                                                // count=1, user mode
  g0[1] = lds_byte_addr;                                     // LDS byte address
  g0[2] = (uint32_t)ga;                                      // global_addr[31:0]
  g0[3] = (uint32_t)((ga >> 32) & 0x01FFFFFFu) | (2u << 30); // global_addr[56:32], type=2
  v8i g1;
  g1[0] = (int)(2u << 16);                                   // data_size = 2 (4 bytes)
  g1[1] = (int)((tensor_elems & 0xFFFFu) << 16);             // tensor_dim0[15:0]
  g1[2] = (int)((tensor_elems >> 16) | (1u << 16));          // tensor_dim0[31:16], tensor_dim1=1
  g1[3] = (int)((tile_elems & 0xFFFFu) << 16);               // tile_dim0
  g1[4] = 0;                                                 // tile_dim1=0 (1-D), tile_dim2=0
  g1[5] = (int)tensor_elems;                                 // tensor_dim0_stride[31:0] (unused, safe)
  g1[6] = 0;
  g1[7] = 0;
  v4i z = {0, 0, 0, 0};
#if defined(__clang_major__) && (__clang_major__ >= 23)
  v8i z8 = {0, 0, 0, 0, 0, 0, 0, 0};
  __builtin_amdgcn_tensor_load_to_lds(g0, g1, z, z, z8, 0);
#else
  __builtin_amdgcn_tensor_load_to_lds(g0, g1, z, z, 0);
#endif
#else
  (void)lds_byte_addr; (void)gptr; (void)tensor_elems; (void)tile_elems;
#endif
}

// ---------------- per-edge SH compute + scatter-accumulate ----------------
// acc layout: [N][9] = {sh1..sh8 sums, count}. sh0 is constant -> handled in finalize.
__device__ __forceinline__ void sh_accum(const float* __restrict__ pos,
                                         float* __restrict__ acc, int s, int d) {
  const float* ps = pos + 3 * s;
  const float* pd = pos + 3 * d;
  float sx = ps[0], sy = ps[1], sz = ps[2];
  float dx = pd[0], dy = pd[1], dz = pd[2];
  float x = dx - sx, y = dy - sy, z = dz - sz;
  float r = sqrtf(x * x + y * y + z * z);
  float inv = 1.0f / (r + 1e-12f);
  x *= inv; y *= inv; z *= inv;
  float* a = acc + 9 * d;
  unsafeAtomicAdd(a + 0, SH_C1 * x);
  unsafeAtomicAdd(a + 1, SH_C1 * y);
  unsafeAtomicAdd(a + 2, SH_C1 * z);
  unsafeAtomicAdd(a + 3, SH_C2A * x * y);
  unsafeAtomicAdd(a + 4, SH_C2A * y * z);
  unsafeAtomicAdd(a + 5, SH_C2B * (3.0f * z * z - 1.0f));
  unsafeAtomicAdd(a + 6, SH_C2A * x * z);
  unsafeAtomicAdd(a + 7, SH_C2C * (x * x - y * y));
  unsafeAtomicAdd(a + 8, 1.0f);
}

// ---------------- main edge kernel (first in file so disasm snippet shows it) ----------------
__global__ __launch_bounds__(BLOCK)
void sh_edge_kernel(const float* __restrict__ pos,
                    const int* __restrict__ src,
                    const int* __restrict__ dst,
                    float* __restrict__ acc,
                    int E, int numChunks) {
#if USE_TDM
  // [buffer][src(0)/dst(1)][wave][elem] : 16 KB LDS, private slice per wave.
  __shared__ int s_idx[2][2][WAVES][CHUNK];
  const int wave = (int)(threadIdx.x >> 5);
  const int lane = (int)(threadIdx.x & 31);
  const int gw0 = (int)blockIdx.x * WAVES + wave;
  const int stride = (int)gridDim.x * WAVES;
  if (gw0 >= numChunks) return;

  const uint32_t as0 = (uint32_t)(uintptr_t)&s_idx[0][0][wave][0];
  const uint32_t as1 = (uint32_t)(uintptr_t)&s_idx[1][0][wave][0];
  const uint32_t ad0 = (uint32_t)(uintptr_t)&s_idx[0][1][wave][0];
  const uint32_t ad1 = (uint32_t)(uintptr_t)&s_idx[1][1][wave][0];

  // prologue: DMA first chunk into buffer 0 (one issue per wave; all lanes active)
  {
    int b0 = gw0 * CHUNK;
    uint32_t rem = (uint32_t)(E - b0);
    tdm_load_1d(as0, src + b0, rem, CHUNK);
    tdm_load_1d(ad0, dst + b0, rem, CHUNK);
  }

  int k = 0;
  for (int c = gw0; c < numChunks; c += stride, ++k) {
    const int p = k & 1;
    const int cn = c + stride;
    if (cn < numChunks) {
      // prefetch next chunk into the other buffer; wait only for current chunk's 2 ops
      int bn = cn * CHUNK;
      uint32_t rem = (uint32_t)(E - bn);
      tdm_load_1d(p ? as0 : as1, src + bn, rem, CHUNK);
      tdm_load_1d(p ? ad0 : ad1, dst + bn, rem, CHUNK);
      __builtin_amdgcn_s_wait_tensorcnt(2);
    } else {
      __builtin_amdgcn_s_wait_tensorcnt(0);
    }

    const int base = c * CHUNK;
    const int lim = E - base;  // > 0 by construction
    if (lim >= CHUNK) {
      // full chunk: no per-edge bounds check
#pragma unroll
      for (int i = 0; i < CHUNK / 32; ++i) {
        const int j = lane + 32 * i;
        sh_accum(pos, acc, s_idx[p][0][wave][j], s_idx[p][1][wave][j]);
      }
    } else {
      // tail chunk
#pragma unroll
      for (int i = 0; i < CHUNK / 32; ++i) {
        const int j = lane + 32 * i;
        if (j < lim) {
          sh_accum(pos, acc, s_idx[p][0][wave][j], s_idx[p][1][wave][j]);
        }
      }
    }
  }
#else
  // Fallback path (no TDM builtin): plain grid-stride over edges.
  (void)numChunks;
  int e = (int)(blockIdx.x * blockDim.x + threadIdx.x);
  int st = (int)(gridDim.x * blockDim.x);
  for (; e < E; e += st) {
    sh_accum(pos, acc, src[e], dst[e]);
  }
#endif
}

// ---------------- helper kernels ----------------
__global__ void zero_f32(float* __restrict__ p, int n) {
  int i = (int)(blockIdx.x * blockDim.x + threadIdx.x);
  int st = (int)(gridDim.x * blockDim.x);
  for (; i < n; i += st) p[i] = 0.0f;
}

__global__ void sh_finalize(const float* __restrict__ acc, float* __restrict__ out, int N) {
  int n = (int)(blockIdx.x * blockDim.x + threadIdx.x);
  if (n >= N) return;
  const float* a = acc + 9 * n;
  float cnt = a[8];
  float inv = 1.0f / fmaxf(cnt, 1.0f);
  float* o = out + 9 * n;
  o[0] = (cnt > 0.0f) ? SH_C0 : 0.0f;  // mean of constant sh0
  o[1] = a[0] * inv;
  o[2] = a[1] * inv;
  o[3] = a[2] * inv;
  o[4] = a[3] * inv;
  o[5] = a[4] * inv;
  o[6] = a[5] * inv;
  o[7] = a[6] * inv;
  o[8] = a[7] * inv;
}

// ---------------- launcher ----------------
extern "C" void kernel_launch(void* const* d_in, const int* in_sizes, int n_in,
                              void* d_out, int out_size, void* d_ws, size_t ws_size,
                              hipStream_t stream) {
  const float* pos = (const float*)d_in[0];
  const int* src = (const int*)d_in[1];
  const int* dst = (const int*)d_in[2];
  const int N = in_sizes[0] / 3;
  const int E = in_sizes[1];
  float* acc = (float*)d_ws;         // [N][9] accumulator (sh1..sh8, count)
  float* out = (float*)d_out;        // [N][9]
  (void)n_in; (void)out_size; (void)ws_size;

  zero_f32<<<512, BLOCK, 0, stream>>>(acc, N * 9);

  const int numChunks = (E + CHUNK - 1) / CHUNK;
  int blocks = 768;  // ~6K waves, a few chunks each so DMA double-buffering overlaps
  const int maxb = (numChunks + WAVES - 1) / WAVES;
  if (blocks > maxb) blocks = maxb;
  sh_edge_kernel<<<blocks, BLOCK, 0, stream>>>(pos, src, dst, acc, E, numChunks);

  sh_finalize<<<(N + BLOCK - 1) / BLOCK, BLOCK, 0, stream>>>(acc, out, N);
}